// point_embedding_76656576299158
// MI455X (gfx1250) — compile-verified
//
#include <hip/hip_runtime.h>

// MI455X / gfx1250 point-embedding kernel.
// Bandwidth-bound (~115 MB traffic @ 23.3 TB/s); the (N,10)x(10,64) projection
// is done with V_WMMA_F32_16X16X4_F32 (full fp32 precision, matrix pipe).
// Bias is folded into the initial WMMA accumulator (D = A*B + C), removing the
// post-matmul v_add_f32 tail seen in the previous round's assembly.

typedef __attribute__((ext_vector_type(2))) float v2f;
typedef __attribute__((ext_vector_type(8))) float v8f;

#define NPTS 65536   // N
#define KNBR 16      // K neighbors
#define HID  64      // hidden
#define FMAXV 3.402823466e+38f

__global__ __launch_bounds__(256) void point_embed_kernel(
    const float* __restrict__ pos,       // (B,3,N)
    const long long* __restrict__ idx,   // (B,N,K) int64
    const float* __restrict__ dist,      // (B,N,K)
    const float* __restrict__ W,         // (10,64)
    const float* __restrict__ bias,      // (64,)
    float* __restrict__ out)             // (B,N,64)
{
    // feature staging: 256 points x 12 floats (10 channels, K-padded to 12)
    __shared__ float feat_lds[256 * 12];

    const int tid     = threadIdx.x;
    const int lane    = tid & 31;
    const int waveId  = tid >> 5;
    const int halfSel = lane >> 4;   // 0 = lanes 0-15, 1 = lanes 16-31
    const int lane16  = lane & 15;

    // one point per lane; blocks never straddle a batch (N % 256 == 0)
    const int g = blockIdx.x * 256 + tid;       // flat (b,n) index
    const int b = g >> 16;                      // g / N
    const int n = g & (NPTS - 1);               // g % N

    const float*     posb  = pos  + (size_t)b * 3 * NPTS;
    const long long* idxp  = idx  + (size_t)g * KNBR;
    const float*     distp = dist + (size_t)g * KNBR;

    // center point
    const float cx = posb[n];
    const float cy = posb[NPTS + n];
    const float cz = posb[2 * NPTS + n];

    // gather-reduce over 16 neighbors (full unroll -> many outstanding loads)
    float mxx = -FMAXV, mxy = -FMAXV, mxz = -FMAXV;
    float mnx =  FMAXV, mny =  FMAXV, mnz =  FMAXV;
    float dmx = -FMAXV;
    #pragma unroll
    for (int k = 0; k < KNBR; ++k) {
        const long long i = __builtin_nontemporal_load(&idxp[k]);  // streamed once
        const float nx = posb[i];                 // L2-resident gathers
        const float ny = posb[NPTS + i];
        const float nz = posb[2 * NPTS + i];
        const float dk = __builtin_nontemporal_load(&distp[k]);    // streamed once
        mxx = fmaxf(mxx, nx); mnx = fminf(mnx, nx);
        mxy = fmaxf(mxy, ny); mny = fminf(mny, ny);
        mxz = fmaxf(mxz, nz); mnz = fminf(mnz, nz);
        dmx = fmaxf(dmx, dk);
    }

    // 10-channel feature, zero-padded K 10->12 so A-fragment reads need no masking
    {
        float* f = feat_lds + tid * 12;
        f[0] = cx;        f[1] = cy;        f[2] = cz;         // max of broadcast
        f[3] = mxx;       f[4] = mxy;       f[5] = mxz;        // max_k neighbor
        f[6] = cx - mnx;  f[7] = cy - mny;  f[8] = cz - mnz;   // max_k (c - n) = c - min_k n
        f[9] = dmx;
        f[10] = 0.0f;     f[11] = 0.0f;
    }

    // B fragments from W (10x64), K padded to 12 with zero rows.
    // Layout (K x N, per C-style half split): VGPR0 = rows {4k, 4k+2}, VGPR1 = rows {4k+1, 4k+3}.
    v2f bfrag[4][3];
    #pragma unroll
    for (int j = 0; j < 4; ++j) {
        #pragma unroll
        for (int k = 0; k < 3; ++k) {
            const int kr  = 4 * k + 2 * halfSel;
            const int col = j * 16 + lane16;
            v2f bv;
            bv.x = (kr     < 10) ? W[kr * HID + col]       : 0.0f;
            bv.y = (kr + 1 < 10) ? W[(kr + 1) * HID + col] : 0.0f;
            bfrag[j][k] = bv;
        }
    }
    // bias for this lane's output column in each column tile; every C element a
    // lane holds shares its column, so the accumulator init is a splat.
    float bias4[4];
    #pragma unroll
    for (int j = 0; j < 4; ++j) bias4[j] = bias[j * 16 + lane16];

    __syncthreads();   // feat visible across lanes (ds writes complete)

    // Each wave: 32 points = two 16-row A tiles; 4 column tiles of 16; 3 K-chunks of 4.
    #pragma unroll
    for (int t = 0; t < 2; ++t) {
        const int   srcT = waveId * 32 + t * 16 + lane16;
        const float* fs  = feat_lds + srcT * 12;
        // A 16x4 layout: lanes 0-15 hold K={4k,4k+1}, lanes 16-31 hold K={4k+2,4k+3}
        v2f a[3];
        #pragma unroll
        for (int k = 0; k < 3; ++k) {
            const int kr = 4 * k + 2 * halfSel;          // even -> 8B aligned ds_load_b64
            a[k] = *(const v2f*)(fs + kr);
        }
        const int tileBase = blockIdx.x * 256 + waveId * 32 + t * 16;
        #pragma unroll
        for (int j = 0; j < 4; ++j) {
            // C initialized with the bias: D = feat@W + b comes out of the MACs.
            v8f c;
            #pragma unroll
            for (int r = 0; r < 8; ++r) c[r] = bias4[j];
            #pragma unroll
            for (int k = 0; k < 3; ++k) {
                // D = A(16x4) x B(4x16) + C : v_wmma_f32_16x16x4_f32
                c = __builtin_amdgcn_wmma_f32_16x16x4_f32(
                        false, a[k], false, bfrag[j][k], (short)0, c, false, false);
            }
            const int col = j * 16 + lane16;
            #pragma unroll
            for (int r = 0; r < 8; ++r) {
                const int row = r + 8 * halfSel;         // C VGPR r: M=r / M=r+8
                const float v = fmaxf(c[r], 0.0f);       // ReLU
                // output is write-once, never re-read -> non-temporal store
                __builtin_nontemporal_store(v, &out[(size_t)(tileBase + row) * HID + col]);
            }
        }
    }
}

extern "C" void kernel_launch(void* const* d_in, const int* in_sizes, int n_in,
                              void* d_out, int out_size, void* d_ws, size_t ws_size,
                              hipStream_t stream) {
    const float*     pos  = (const float*)d_in[0];
    const long long* idx  = (const long long*)d_in[1];
    const float*     dist = (const float*)d_in[2];
    const float*     W    = (const float*)d_in[3];
    const float*     bias = (const float*)d_in[4];
    float*           out  = (float*)d_out;

    const int B = 4;
    const int totalPts = B * NPTS;               // 262144
    dim3 grid(totalPts / 256);                   // 1024 blocks, all lanes active (EXEC all-1s)
    point_embed_kernel<<<grid, 256, 0, stream>>>(pos, idx, dist, W, bias, out);
}